// Encoder_48533130445491
// MI455X (gfx1250) — compile-verified
//
#include <hip/hip_runtime.h>

#define N_X     50000
#define N_MOTIF 10000
#define N_TOT   (N_X + N_MOTIF)   // 60000, divisible by 16
#define NFEAT   512
#define NHID    256
#define PAD_N   65536
#define SCAN_B  256

typedef __attribute__((ext_vector_type(2))) float v2f;
typedef __attribute__((ext_vector_type(8))) float v8f;

// ---------------------------------------------------------------------------
// Dense GEMM: C[M x 256] = A[M x K] * B[K x 256], fp32 V_WMMA_F32_16X16X4_F32.
// One wave computes a 16x64 strip (4 accumulators share one A fetch -> 4 WMMAs
// per A load); 4 waves/block cover all 256 columns of a 16-row tile, so A is
// streamed from HBM exactly once. A layout (16x4 f32): lanes 0-15 K={k,k+1},
// lanes 16-31 K={k+2,k+3}; B symmetric; C/D: VGPR j = row j+8*half, col lane&15.
// ---------------------------------------------------------------------------
__global__ __launch_bounds__(128) void gemm_wmma_f32(
    const float* __restrict__ A, const float* __restrict__ B,
    float* __restrict__ C, int K) {
  const int lane = threadIdx.x & 31;
  const int wave = threadIdx.x >> 5;
  const int row0 = blockIdx.x << 4;   // 16-row tile
  const int col0 = wave << 6;         // 64-col strip per wave
  const int m    = lane & 15;
  const int half = lane >> 4;

  const float* __restrict__ arow  = A + (size_t)(row0 + m) * K + (half << 1);
  const float* __restrict__ bbase = B + (size_t)(half << 1) * NHID + col0 + m;

  v8f acc[4] = {{}, {}, {}, {}};
  for (int k = 0; k < K; k += 4) {
    v2f a;
    a.x = arow[0];
    a.y = arow[1];
    #pragma unroll
    for (int t = 0; t < 4; ++t) {
      v2f b;
      b.x = bbase[t * 16];
      b.y = bbase[t * 16 + NHID];
      acc[t] = __builtin_amdgcn_wmma_f32_16x16x4_f32(
          false, a, false, b, (short)0, acc[t], false, false);
    }
    arow  += 4;
    bbase += 4 * NHID;
  }
  #pragma unroll
  for (int t = 0; t < 4; ++t) {
    float* __restrict__ crow =
        C + (size_t)(row0 + (half << 3)) * NHID + col0 + t * 16 + m;
    #pragma unroll
    for (int j = 0; j < 8; ++j)
      crow[(size_t)j * NHID] = acc[t][j];
  }
}

// ------------------------- CSR construction (once) -------------------------
__global__ __launch_bounds__(256) void hist_rows(
    const int* __restrict__ rows, int* __restrict__ cnt, int nE) {
  int i = blockIdx.x * blockDim.x + threadIdx.x;
  if (i < nE) atomicAdd(&cnt[rows[i]], 1);
}

__global__ __launch_bounds__(SCAN_B) void scan_blocks(
    const int* __restrict__ cnt, int* __restrict__ excl,
    int* __restrict__ bsum, int n) {
  __shared__ int sm[SCAN_B];
  const int i = blockIdx.x * SCAN_B + threadIdx.x;
  const int v = (i < n) ? cnt[i] : 0;
  sm[threadIdx.x] = v;
  __syncthreads();
  for (int off = 1; off < SCAN_B; off <<= 1) {   // Hillis-Steele inclusive
    int t = (threadIdx.x >= off) ? sm[threadIdx.x - off] : 0;
    __syncthreads();
    sm[threadIdx.x] += t;
    __syncthreads();
  }
  if (i < n) excl[i] = sm[threadIdx.x] - v;      // exclusive within block
  if (threadIdx.x == SCAN_B - 1) bsum[blockIdx.x] = sm[SCAN_B - 1];
}

__global__ void scan_partials(int* __restrict__ bsum, int nb) {
  if (threadIdx.x == 0) {
    int run = 0;
    for (int i = 0; i < nb; ++i) { int t = bsum[i]; bsum[i] = run; run += t; }
  }
}

__global__ __launch_bounds__(256) void add_offsets(
    const int* __restrict__ excl, const int* __restrict__ bsum,
    int* __restrict__ rowptr, int n, int nE) {
  int i = blockIdx.x * blockDim.x + threadIdx.x;
  if (i < n) rowptr[i] = excl[i] + bsum[i >> 8];
  if (i == 0) rowptr[n] = nE;
}

__global__ __launch_bounds__(256) void fill_csr(
    const int* __restrict__ rows, const int* __restrict__ cols,
    const float* __restrict__ vals, const int* __restrict__ rowptr,
    int* __restrict__ fillc, int* __restrict__ cols_s,
    float* __restrict__ vals_s, int nE) {
  int i = blockIdx.x * blockDim.x + threadIdx.x;
  if (i >= nE) return;
  const int r = rows[i];
  const int idx = rowptr[r] + atomicAdd(&fillc[r], 1);
  cols_s[idx] = cols[i];
  vals_s[idx] = vals[i];
}

// ---------------------------------------------------------------------------
// Gather-side SpMM + bias + ReLU: hout[r] = relu(sum_e val_e*sup[col_e] + b).
// One wave per row; lane owns 8 contiguous feats -> 2x b128 gathers per edge
// from the L2-resident support matrix. No atomics.
// ---------------------------------------------------------------------------
__global__ __launch_bounds__(256) void spmm_gather_bias_relu(
    const int* __restrict__ rowptr, const int* __restrict__ cols_s,
    const float* __restrict__ vals_s, const float* __restrict__ sup,
    const float* __restrict__ bias, float* __restrict__ hout, int nRows) {
  const int r = blockIdx.x * (blockDim.x >> 5) + (threadIdx.x >> 5);
  if (r >= nRows) return;
  const int lane = threadIdx.x & 31;
  const int f0 = lane << 3;   // 8 contiguous features per lane

  float4 acc0 = {0.f, 0.f, 0.f, 0.f};
  float4 acc1 = {0.f, 0.f, 0.f, 0.f};
  const int e1 = rowptr[r + 1];
  for (int e = rowptr[r]; e < e1; ++e) {
    const int   c = cols_s[e];
    const float v = vals_s[e];
    const float4* __restrict__ s =
        (const float4*)(sup + (size_t)c * NHID + f0);
    const float4 s0 = s[0], s1 = s[1];
    acc0.x += v * s0.x; acc0.y += v * s0.y; acc0.z += v * s0.z; acc0.w += v * s0.w;
    acc1.x += v * s1.x; acc1.y += v * s1.y; acc1.z += v * s1.z; acc1.w += v * s1.w;
  }
  const float4* __restrict__ bp = (const float4*)(bias + f0);
  const float4 b0 = bp[0], b1 = bp[1];
  float4 o0, o1;
  o0.x = fmaxf(acc0.x + b0.x, 0.f); o0.y = fmaxf(acc0.y + b0.y, 0.f);
  o0.z = fmaxf(acc0.z + b0.z, 0.f); o0.w = fmaxf(acc0.w + b0.w, 0.f);
  o1.x = fmaxf(acc1.x + b1.x, 0.f); o1.y = fmaxf(acc1.y + b1.y, 0.f);
  o1.z = fmaxf(acc1.z + b1.z, 0.f); o1.w = fmaxf(acc1.w + b1.w, 0.f);
  float4* __restrict__ d = (float4*)(hout + (size_t)r * NHID + f0);
  d[0] = o0;
  d[1] = o1;
}

// out[pos[r]][:] = h[r][:]  (d_out pre-zeroed); 2x b128 per lane, 1 wave/row
__global__ __launch_bounds__(256) void scatter_rows(
    const float* __restrict__ h, const int* __restrict__ pos,
    float* __restrict__ out) {
  const int r = blockIdx.x * (blockDim.x >> 5) + (threadIdx.x >> 5);
  if (r >= N_X) return;
  const int lane = threadIdx.x & 31;
  const int f0 = lane << 3;
  const int p = pos[r];
  const float4* __restrict__ s = (const float4*)(h + (size_t)r * NHID + f0);
  float4* __restrict__ d = (float4*)(out + (size_t)p * NHID + f0);
  d[0] = s[0];
  d[1] = s[1];
}

extern "C" void kernel_launch(void* const* d_in, const int* in_sizes, int n_in,
                              void* d_out, int out_size, void* d_ws, size_t ws_size,
                              hipStream_t stream) {
  const float* x     = (const float*)d_in[0];
  const float* motif = (const float*)d_in[1];
  const int*   rows  = (const int*)d_in[2];
  const int*   cols  = (const int*)d_in[3];
  const float* vals  = (const float*)d_in[4];
  const int*   pos   = (const int*)d_in[5];
  const float* w1 = (const float*)d_in[7];
  const float* b1 = (const float*)d_in[8];
  const float* w2 = (const float*)d_in[9];
  const float* b2 = (const float*)d_in[10];
  const float* w3 = (const float*)d_in[11];
  const float* b3 = (const float*)d_in[12];

  const int E    = in_sizes[2];
  const int NBLK = (N_TOT + SCAN_B - 1) / SCAN_B;   // 235

  // ---- workspace layout (fp32 regions first: 16B-aligned float4 access) ----
  float* H0  = (float*)d_ws;                       // N_TOT x 512  (also reused as h2)
  float* SUP = H0  + (size_t)N_TOT * NFEAT;        // N_TOT x 256
  float* HBa = SUP + (size_t)N_TOT * NHID;         // N_TOT x 256  (h1 / h3)
  int*   cnt    = (int*)(HBa + (size_t)N_TOT * NHID);
  int*   excl   = cnt    + N_TOT;
  int*   rowptr = excl   + N_TOT;                  // N_TOT+1
  int*   fillc  = rowptr + N_TOT + 1;
  int*   bsum   = fillc  + N_TOT;                  // NBLK
  int*   cols_s = bsum   + ((NBLK + 63) & ~63);
  float* vals_s = (float*)(cols_s + E);
  float* HBb = H0;                                 // layer-2 output reuses H0

  // ---- h0 = concat(x, motif_emb) ----
  hipMemcpyAsync(H0, x, (size_t)N_X * NFEAT * sizeof(float),
                 hipMemcpyDeviceToDevice, stream);
  hipMemcpyAsync(H0 + (size_t)N_X * NFEAT, motif,
                 (size_t)N_MOTIF * NFEAT * sizeof(float),
                 hipMemcpyDeviceToDevice, stream);

  // ---- CSR build (graph is static across the 3 layers) ----
  hipMemsetAsync(cnt,   0, (size_t)N_TOT * sizeof(int), stream);
  hipMemsetAsync(fillc, 0, (size_t)N_TOT * sizeof(int), stream);
  const int eblk = (E + 255) / 256;
  hist_rows<<<eblk, 256, 0, stream>>>(rows, cnt, E);
  scan_blocks<<<NBLK, SCAN_B, 0, stream>>>(cnt, excl, bsum, N_TOT);
  scan_partials<<<1, 32, 0, stream>>>(bsum, NBLK);
  add_offsets<<<(N_TOT + 255) / 256, 256, 0, stream>>>(excl, bsum, rowptr, N_TOT, E);
  fill_csr<<<eblk, 256, 0, stream>>>(rows, cols, vals, rowptr, fillc, cols_s, vals_s, E);

  const dim3 gblk(128);
  const dim3 ggrid(N_TOT / 16);                 // 3750 blocks, 16x256 out each
  const int  rblk = (N_TOT + 7) / 8;            // 8 waves (rows) per block

  // ---- Layer 1 ----
  gemm_wmma_f32<<<ggrid, gblk, 0, stream>>>(H0, w1, SUP, NFEAT);
  spmm_gather_bias_relu<<<rblk, 256, 0, stream>>>(rowptr, cols_s, vals_s, SUP, b1, HBa, N_TOT);
  // ---- Layer 2 ----
  gemm_wmma_f32<<<ggrid, gblk, 0, stream>>>(HBa, w2, SUP, NHID);
  spmm_gather_bias_relu<<<rblk, 256, 0, stream>>>(rowptr, cols_s, vals_s, SUP, b2, HBb, N_TOT);
  // ---- Layer 3 ----
  gemm_wmma_f32<<<ggrid, gblk, 0, stream>>>(HBb, w3, SUP, NHID);
  spmm_gather_bias_relu<<<rblk, 256, 0, stream>>>(rowptr, cols_s, vals_s, SUP, b3, HBa, N_TOT);

  // ---- Output: zero 65536x256 then scatter first N_X rows ----
  hipMemsetAsync(d_out, 0, (size_t)PAD_N * NHID * sizeof(float), stream);
  scatter_rows<<<(N_X + 7) / 8, 256, 0, stream>>>(HBa, pos, (float*)d_out);
}